// SpaceCarverGridSamplerModule_67757404062167
// MI455X (gfx1250) — compile-verified
//
#include <hip/hip_runtime.h>

// SpaceCarver grid sampler (nearest + 3x3 fix-search, Zeros padding,
// align_corners=True) for MI455X / gfx1250.
//
// CDNA5-specific paths used:
//  - global_load_async_to_lds_b128 + s_wait_asynccnt double-buffered staging
//    of the streamed grid (ASYNCcnt pipeline, no VGPR long-latency waits)
//  - non-temporal output stores so the 8MB depth image stays L2-resident
//    while 96MB of grid/out streams past it.

typedef float v2f __attribute__((ext_vector_type(2)));
typedef float v4f __attribute__((ext_vector_type(4)));

#define TPB   256   // threads per block (8 wave32)
#define PPT   2     // pixels per thread per iteration (one b128 of grid)
#define ITERS 8     // chunks per block

__device__ __forceinline__ float sample_fix(const float* __restrict__ img,
                                            float gx, float gy, int H, int W) {
  // unnormalize, align_corners=True; same f32 op order as reference
  float x = (gx + 1.0f) * 0.5f * (float)(W - 1);
  float y = (gy + 1.0f) * 0.5f * (float)(H - 1);
  // jnp.round == round-half-to-even == v_rndne_f32
  int ix = (int)__builtin_rintf(x);
  int iy = (int)__builtin_rintf(y);

  // center candidate (priority 0)
  bool cin = (ix >= 0) && (ix < W) && (iy >= 0) && (iy < H);
  int cx = min(max(ix, 0), W - 1);
  int cy = min(max(iy, 0), H - 1);
  float vc = img[cy * W + cx];
  bool cvalid = cin && (vc != 0.0f);
  float res = cvalid ? vc : 0.0f;

  // ring candidates, row-major priority; lanes with a valid center are
  // EXEC-masked off here (no extra gather traffic for them).
  if (!cvalid) {
    const int dys[8] = {-1, -1, -1, 0, 0, 1, 1, 1};
    const int dxs[8] = {-1,  0,  1,-1, 1,-1, 0, 1};
    float sel = 0.0f;
    bool  any = false;
#pragma unroll
    for (int k = 0; k < 8; ++k) {
      int yk = iy + dys[k];
      int xk = ix + dxs[k];
      bool inb = (xk >= 0) && (xk < W) && (yk >= 0) && (yk < H);
      int yc = min(max(yk, 0), H - 1);
      int xc = min(max(xk, 0), W - 1);
      float v = img[yc * W + xc];      // clamped address, branch-free
      bool valid = inb && (v != 0.0f);
      sel = (valid && !any) ? v : sel; // first-valid priority select
      any = any || valid;
    }
    res = sel;                          // stays 0.0f if none valid
  }
  return res;
}

__global__ __launch_bounds__(TPB) void
spacecarver_sampler_kernel(const float* __restrict__ depth,
                           const float* __restrict__ grid,
                           float* __restrict__ out,
                           int H, int W, int bshift, int N) {
  __shared__ v4f sg[2][TPB];  // double-buffered grid stage: 2 pixels/lane
  const int tid  = threadIdx.x;
  const int HW   = H * W;
  const int base = (int)blockIdx.x * (ITERS * TPB * PPT);

  // LDS byte offsets for this lane's two staging slots.
  // Generic LDS pointers on gfx1250 are {SHARED_BASE, 32'offset}; the low
  // 32 bits are exactly the LDS byte address the async op wants.
  unsigned lds0 = (unsigned)(unsigned long long)&sg[0][tid];
  unsigned lds1 = (unsigned)(unsigned long long)&sg[1][tid];

  // Prologue: stage chunk 0 into buffer 0 (ASYNCcnt++)
  {
    int p = min(base + tid * PPT, N - PPT);     // clamp keeps address in-bounds
    const float* gp = grid + (size_t)p * 2;
    asm volatile("global_load_async_to_lds_b128 %0, %1, off"
                 :: "v"(lds0), "v"(gp) : "memory");
  }

  for (int it = 0; it < ITERS; ++it) {
    const int chunk = base + it * (TPB * PPT);

    if (it + 1 < ITERS) {
      // issue prefetch of chunk it+1 into the other buffer, then wait for
      // the older copy (async loads complete in order) -> overlap copy/compute
      int pn = min(chunk + TPB * PPT + tid * PPT, N - PPT);
      const float* gp = grid + (size_t)pn * 2;
      unsigned lnext = ((it + 1) & 1) ? lds1 : lds0;
      asm volatile("global_load_async_to_lds_b128 %0, %1, off"
                   :: "v"(lnext), "v"(gp) : "memory");
      asm volatile("s_wait_asynccnt 0x1" ::: "memory");
    } else {
      asm volatile("s_wait_asynccnt 0x0" ::: "memory");
    }

    // each lane consumes exactly the 16B it staged -> no barrier needed
    v4f g = sg[it & 1][tid];

    int p0 = chunk + tid * PPT;
    if (p0 + PPT <= N) {
      const float* img = depth + (size_t)(p0 >> bshift) * HW;  // batch slice
      v2f r;
      r.x = sample_fix(img, g.x, g.y, H, W);
      r.y = sample_fix(img, g.z, g.w, H, W);
      // NT store: don't let the 32MB output stream evict L2-resident depth
      __builtin_nontemporal_store(r, (v2f*)(out + p0));
    }
  }
}

extern "C" void kernel_launch(void* const* d_in, const int* in_sizes, int n_in,
                              void* d_out, int out_size, void* d_ws, size_t ws_size,
                              hipStream_t stream) {
  const float* depth = (const float*)d_in[0];  // (B,1,H,W) f32
  const float* grid  = (const float*)d_in[1];  // (B,Ho,Wo,2) f32
  float* out = (float*)d_out;                  // (B,1,Ho,Wo) f32

  const int H = 512, W = 512;
  const int B = in_sizes[0] / (H * W);         // C==1
  const int N = out_size;                      // B*Ho*Wo
  const int hoWo = N / B;                      // power of two (1024*1024)

  int bshift = 0;
  while ((1 << bshift) < hoWo) ++bshift;       // p >> bshift == batch index

  const int perBlock = TPB * PPT * ITERS;      // 4096 pixels per block
  const int blocks = (N + perBlock - 1) / perBlock;  // 2048 for this shape

  spacecarver_sampler_kernel<<<blocks, TPB, 0, stream>>>(
      depth, grid, out, H, W, bshift, N);
}